// MultilevelCropResize_20169166422200
// MI455X (gfx1250) — compile-verified
//
#include <hip/hip_runtime.h>
#include <cstdint>

// Multilevel crop-and-resize (ROI-align), B=2, N=512, C=256, out 7x7, levels 2..6.
// Memory-bound gather: ~205MB tap reads (L2-resident, 192MB L2) + 51MB NT writes
// -> ~11us roofline @ 23.3TB/s. One wave32 per (b,n,p,q) sample; the 4 tap rows
// (4x1KB) are staged via CDNA5 async global->LDS (global_load_async_to_lds_b128,
// s_wait_asynccnt), then blended with vector FMAs and stored non-temporally.

#define OUT_S   7
#define NCH     256
#define NBOX    512
#define WAVES_PER_BLOCK 8
#define TOTAL_WAVES (2 * NBOX * OUT_S * OUT_S)   // 50176

#if defined(__has_builtin)
#  if __has_builtin(__builtin_amdgcn_global_load_async_to_lds_b128)
#    define HAVE_ASYNC_LDS 1
#  endif
#  if __has_builtin(__builtin_amdgcn_s_wait_asynccnt)
#    define HAVE_WAIT_ASYNC 1
#  endif
#endif

typedef int   v4i __attribute__((ext_vector_type(4)));
typedef float v4f __attribute__((ext_vector_type(4)));

__device__ __forceinline__ void cp16_async(const float* g, float* l) {
#if defined(HAVE_ASYNC_LDS)
  __builtin_amdgcn_global_load_async_to_lds_b128((v4i*)g, (v4i*)l, 0, 0);
#else
  (void)g; (void)l;
#endif
}

__device__ __forceinline__ void wait_async_zero() {
#if defined(HAVE_WAIT_ASYNC)
  __builtin_amdgcn_s_wait_asynccnt(0);
  asm volatile("" ::: "memory");
#else
  asm volatile("s_wait_asynccnt 0" ::: "memory");
#endif
}

__global__ __launch_bounds__(256) void MultilevelCropResize_kernel(
    const float* __restrict__ f2, const float* __restrict__ f3,
    const float* __restrict__ f4, const float* __restrict__ f5,
    const float* __restrict__ f6, const float* __restrict__ boxes,
    float* __restrict__ out)
{
  __shared__ alignas(16) float smem[WAVES_PER_BLOCK][4 * NCH];  // 32 KB / block

  const unsigned tid   = threadIdx.x;
  const unsigned lane  = tid & 31u;
  const unsigned wslot = tid >> 5;
  unsigned s = blockIdx.x * WAVES_PER_BLOCK + wslot;   // grid sized exactly: no tail

  // decompose s -> (b, n, p, q)  (unsigned: cheaper magic-div sequences)
  const unsigned q = s % OUT_S;  s /= OUT_S;
  const unsigned p = s % OUT_S;  s /= OUT_S;
  const unsigned n = s & (NBOX - 1);
  const unsigned b = s >> 9;

  // ---- box -> level & sampling weights (wave-uniform VALU math) ----
  const float* bx = boxes + ((unsigned long)b * NBOX + n) * 4u;
  const float y_lo = bx[0], x_lo = bx[1], y_hi = bx[2], x_hi = bx[3];
  const float bh = y_hi - y_lo;
  const float bw = x_hi - x_lo;
  const float area_sqrt = sqrtf(bh * bw);
  // level = floor(log(a/224)/log(2)) + 4, clip [2,6]
  int level = (int)(floorf(logf(area_sqrt * (1.0f / 224.0f)) *
                           1.44269504088896340736f) + 4.0f);
  level = level < 2 ? 2 : (level > 6 ? 6 : level);
  const int   lvl       = level - 2;
  const int   hw        = 256 >> lvl;
  const float bnd       = (float)(hw - 1);
  const float inv_scale = exp2f(-(float)level);

  const float gy = y_lo * inv_scale + ((float)p + 0.5f) * (1.0f / 7.0f) * (bh * inv_scale);
  const float gx = x_lo * inv_scale + ((float)q + 0.5f) * (1.0f / 7.0f) * (bw * inv_scale);
  const float yf = fmaxf(floorf(gy), 0.0f);
  const float xf = fmaxf(floorf(gx), 0.0f);
  const float ly = gy - yf;
  const float lx = gx - xf;
  const int yA = (int)fminf(yf,        bnd);
  const int yB = (int)fminf(yf + 1.0f, bnd);
  const int xA = (int)fminf(xf,        bnd);
  const int xB = (int)fminf(xf + 1.0f, bnd);
  const float w00 = (1.0f - ly) * (1.0f - lx);
  const float w01 = (1.0f - ly) * lx;
  const float w10 = ly * (1.0f - lx);
  const float w11 = ly * lx;

  const float* feat = (lvl == 0) ? f2 : (lvl == 1) ? f3 :
                      (lvl == 2) ? f4 : (lvl == 3) ? f5 : f6;
  const long rowA = ((long)b * hw + yA) * hw;
  const long rowB = ((long)b * hw + yB) * hw;
  const float* r0 = feat + (rowA + xA) * (long)NCH;
  const float* r1 = feat + (rowA + xB) * (long)NCH;
  const float* r2 = feat + (rowB + xA) * (long)NCH;
  const float* r3 = feat + (rowB + xB) * (long)NCH;

  const unsigned lo = lane * 4u;  // float offset; lanes cover [0,128) and [128,256)
  v4f v0a, v0b, v1a, v1b, v2a, v2b, v3a, v3b;

#if defined(HAVE_ASYNC_LDS)
  // ---- CDNA5 async global->LDS staging: per-lane 16B, 2 halves per tap row ----
  float* ls = smem[wslot];
  cp16_async(r0 + lo,       ls + 0 * NCH + lo);
  cp16_async(r0 + 128 + lo, ls + 0 * NCH + 128 + lo);
  cp16_async(r1 + lo,       ls + 1 * NCH + lo);
  cp16_async(r1 + 128 + lo, ls + 1 * NCH + 128 + lo);
  cp16_async(r2 + lo,       ls + 2 * NCH + lo);
  cp16_async(r2 + 128 + lo, ls + 2 * NCH + 128 + lo);
  cp16_async(r3 + lo,       ls + 3 * NCH + lo);
  cp16_async(r3 + 128 + lo, ls + 3 * NCH + 128 + lo);
  wait_async_zero();
  v0a = *reinterpret_cast<const v4f*>(ls + 0 * NCH + lo);
  v0b = *reinterpret_cast<const v4f*>(ls + 0 * NCH + 128 + lo);
  v1a = *reinterpret_cast<const v4f*>(ls + 1 * NCH + lo);
  v1b = *reinterpret_cast<const v4f*>(ls + 1 * NCH + 128 + lo);
  v2a = *reinterpret_cast<const v4f*>(ls + 2 * NCH + lo);
  v2b = *reinterpret_cast<const v4f*>(ls + 2 * NCH + 128 + lo);
  v3a = *reinterpret_cast<const v4f*>(ls + 3 * NCH + lo);
  v3b = *reinterpret_cast<const v4f*>(ls + 3 * NCH + 128 + lo);
#else
  // ---- fallback: direct coalesced B128 loads ----
  v0a = *reinterpret_cast<const v4f*>(r0 + lo);
  v0b = *reinterpret_cast<const v4f*>(r0 + 128 + lo);
  v1a = *reinterpret_cast<const v4f*>(r1 + lo);
  v1b = *reinterpret_cast<const v4f*>(r1 + 128 + lo);
  v2a = *reinterpret_cast<const v4f*>(r2 + lo);
  v2b = *reinterpret_cast<const v4f*>(r2 + 128 + lo);
  v3a = *reinterpret_cast<const v4f*>(r3 + lo);
  v3b = *reinterpret_cast<const v4f*>(r3 + 128 + lo);
#endif

  v4f accA = w00 * v0a + w01 * v1a + w10 * v2a + w11 * v3a;
  v4f accB = w00 * v0b + w01 * v1b + w10 * v2b + w11 * v3b;

  // Output is write-once / never re-read: non-temporal stores keep the 179MB of
  // feature maps resident in the 192MB L2 instead of evicting them.
  float* o = out + ((((unsigned long)b * NBOX + n) * (OUT_S * OUT_S)) +
                    p * OUT_S + q) * (unsigned long)NCH;
  __builtin_nontemporal_store(accA, reinterpret_cast<v4f*>(o + lo));
  __builtin_nontemporal_store(accB, reinterpret_cast<v4f*>(o + 128 + lo));
}

extern "C" void kernel_launch(void* const* d_in, const int* in_sizes, int n_in,
                              void* d_out, int out_size, void* d_ws, size_t ws_size,
                              hipStream_t stream) {
  const float* f2    = (const float*)d_in[0];
  const float* f3    = (const float*)d_in[1];
  const float* f4    = (const float*)d_in[2];
  const float* f5    = (const float*)d_in[3];
  const float* f6    = (const float*)d_in[4];
  const float* boxes = (const float*)d_in[5];
  float* out = (float*)d_out;

  const int blocks = TOTAL_WAVES / WAVES_PER_BLOCK;   // 6272, exact
  hipLaunchKernelGGL(MultilevelCropResize_kernel, dim3(blocks), dim3(256), 0, stream,
                     f2, f3, f4, f5, f6, boxes, out);
}